// _Whitening_63075889709430
// MI455X (gfx1250) — compile-verified
//
#include <hip/hip_runtime.h>
#include <math.h>

// ---------------------------------------------------------------------------
// Group whitening (ZCA via Cholesky) for x: (32, 256, 56, 56) fp32.
//   Pass 1: fused per-channel sums + per-group 16x16 Gram (uncentered) using
//           V_WMMA_F32_16X16X4_F32. Covariance recovered as (XX^T - n mu mu^T)/(n-1).
//   Pass 2: per-group Cholesky + lower-triangular inverse (tiny, 16 threads).
//   Pass 3: whitening GEMM out = W*x - W*mu, bias folded into WMMA C operand.
// Memory-bound: ~3 x 103MB traffic => ~13us at 23.3 TB/s. WMMA f32 16x16x4
// matches both GEMM shapes exactly.
// ---------------------------------------------------------------------------

typedef __attribute__((ext_vector_type(2))) float v2f;
typedef __attribute__((ext_vector_type(8))) float v8f;

#define GS 16          // group size (channels per group)
#define NG 16          // number of groups
#define NFEA 256       // channels
#define HWN 3136       // 56*56 (columns per image per channel)
#define MB 32          // batch
#define IMG_STRIDE (NFEA * HWN)   // 802816 floats per image
#define NTOT (MB * HWN)           // 100352 samples per channel

// workspace layout (floats):
//   gram : NG*256 = 4096 @ 0      (per-group uncentered Gram, row-major)
//   sums : 256          @ 4096    (per-channel sums)
//   W    : NG*256 = 4096 @ 4352   (per-group inv-chol, row-major, lower-tri)
//   wmu  : 256          @ 8448    (per-group W*mu bias)

__global__ void k_zero(float* __restrict__ ws, int n) {
    int i = blockIdx.x * blockDim.x + threadIdx.x;
    if (i < n) ws[i] = 0.0f;
}

// ---- Pass 1: Gram + channel sums. Block = (group g, image m). 8 waves. ----
// A-fragment layout for WMMA_F32_16x16x4 (lane L, VGPR v):
//   X[row = L%16][k0 + v + 2*(L/16)]  -> one float2 load per lane per K=4 step.
// The B-fragment (X^T, 4x16 row-striped) has the SAME lane/VGPR placement,
// so the same registers feed both operands.
__global__ __launch_bounds__(256) void k_gram(const float* __restrict__ x,
                                              float* __restrict__ gram,
                                              float* __restrict__ sums) {
    __shared__ float lgram[256];
    __shared__ float lsum[16];

    const int tid  = threadIdx.x;
    const int wave = tid >> 5;
    const int lane = tid & 31;
    const int g    = blockIdx.x & 15;
    const int m    = blockIdx.x >> 4;

    lgram[tid] = 0.0f;
    if (tid < 16) lsum[tid] = 0.0f;
    __syncthreads();

    const int row = lane & 15;
    const int hi  = lane >> 4;
    const float* base = x + (size_t)m * IMG_STRIDE
                          + (size_t)(g * GS + row) * HWN
                          + 2 * hi;
    const int k0 = wave * (HWN / 8);          // 392 columns per wave

    v8f acc0 = {};
    v8f acc1 = {};
    float s = 0.0f;

    for (int kk = 0; kk < HWN / 8; kk += 8) { // 49 iters x 2 WMMA steps
        v2f a0 = *(const v2f*)(base + k0 + kk);
        v2f a1 = *(const v2f*)(base + k0 + kk + 4);
        s += a0.x + a0.y + a1.x + a1.y;       // channel-sum side product
        acc0 = __builtin_amdgcn_wmma_f32_16x16x4_f32(false, a0, false, a0,
                                                     (short)0, acc0, false, false);
        acc1 = __builtin_amdgcn_wmma_f32_16x16x4_f32(false, a1, false, a1,
                                                     (short)0, acc1, false, false);
    }

    // C/D layout: VGPR r, lane L -> (M = r + 8*(L/16), N = L%16)
#pragma unroll
    for (int r = 0; r < 8; r++)
        atomicAdd(&lgram[(r + 8 * hi) * 16 + row], acc0[r] + acc1[r]);
    atomicAdd(&lsum[row], s);
    __syncthreads();

    atomicAdd(&gram[g * 256 + tid], lgram[tid]);
    if (tid < 16) atomicAdd(&sums[g * 16 + tid], lsum[tid]);
}

// ---- Pass 2: sigma -> chol -> inv(T) -> bias. One thread per group. -------
__global__ void k_chol(const float* __restrict__ gram,
                       const float* __restrict__ sums,
                       float* __restrict__ W,
                       float* __restrict__ wmu) {
    __shared__ float Sh[16 * 257];   // padded slabs: stride 257 avoids bank clash
    __shared__ float Wh[16 * 257];
    const int g = threadIdx.x;
    if (g >= 16) return;
    float* Sg = &Sh[g * 257];
    float* Wg = &Wh[g * 257];

    const float nf = (float)NTOT;
    float mu[16];
    for (int j = 0; j < 16; j++) mu[j] = sums[g * 16 + j] / nf;

    const float inv_nm1 = 1.0f / (nf - 1.0f);
    for (int i = 0; i < 16; i++)
        for (int j = 0; j < 16; j++) {
            float sh = (gram[g * 256 + i * 16 + j] - nf * mu[i] * mu[j]) * inv_nm1;
            Sg[i * 16 + j] = (1.0f - 1e-6f) * sh + ((i == j) ? 1e-6f : 0.0f);
        }

    // in-place lower Cholesky
    for (int k = 0; k < 16; k++) {
        float d = sqrtf(Sg[k * 16 + k]);
        Sg[k * 16 + k] = d;
        float inv = 1.0f / d;
        for (int i = k + 1; i < 16; i++) Sg[i * 16 + k] *= inv;
        for (int j = k + 1; j < 16; j++) {
            float f = Sg[j * 16 + k];
            for (int i = j; i < 16; i++) Sg[i * 16 + j] -= Sg[i * 16 + k] * f;
        }
    }

    // W = inv(T), lower triangular
    for (int j = 0; j < 16; j++) {
        for (int i = 0; i < j; i++) Wg[i * 16 + j] = 0.0f;
        Wg[j * 16 + j] = 1.0f / Sg[j * 16 + j];
        for (int i = j + 1; i < 16; i++) {
            float acc = 0.0f;
            for (int k = j; k < i; k++) acc += Sg[i * 16 + k] * Wg[k * 16 + j];
            Wg[i * 16 + j] = -acc / Sg[i * 16 + i];
        }
    }

    for (int t = 0; t < 256; t++) W[g * 256 + t] = Wg[t];
    for (int i = 0; i < 16; i++) {
        float b = 0.0f;
        for (int j = 0; j < 16; j++) b += Wg[i * 16 + j] * mu[j];
        wmu[g * 16 + i] = b;
    }
}

// ---- Pass 3: out = W * x - W*mu via WMMA, bias preloaded into C. ----------
__global__ __launch_bounds__(256) void k_whiten(const float* __restrict__ x,
                                                const float* __restrict__ W,
                                                const float* __restrict__ wmu,
                                                float* __restrict__ out) {
    const int tid  = threadIdx.x;
    const int wave = tid >> 5;
    const int lane = tid & 31;
    const int g    = blockIdx.x & 15;
    const int m    = blockIdx.x >> 4;
    const int col  = lane & 15;
    const int hi   = lane >> 4;

    // A fragments: W[g] split into four 16x4 K-chunks (float2 per lane each)
    const float* Wg = W + g * 256;
    v2f a[4];
#pragma unroll
    for (int c = 0; c < 4; c++)
        a[c] = *(const v2f*)(Wg + col * 16 + c * 4 + 2 * hi);

    // bias fragment in C/D layout: lane holds rows M = r + 8*hi
    v8f cbias;
#pragma unroll
    for (int r = 0; r < 8; r++)
        cbias[r] = -wmu[g * 16 + r + 8 * hi];

    const float* xg = x   + (size_t)m * IMG_STRIDE + (size_t)g * GS * HWN;
    float*       og = out + (size_t)m * IMG_STRIDE + (size_t)g * GS * HWN;

    // 196 column-tiles of 16; wave-uniform loop keeps EXEC all-ones for WMMA
    for (int t = wave; t < HWN / 16; t += 8) {
        const int hw0 = t * 16;
        const float* xb = xg + hw0 + col;
        v8f acc = cbias;
#pragma unroll
        for (int c = 0; c < 4; c++) {
            // B fragment (4x16): lane L, VGPR v = X[ch = 4c+v+2*hi][hw0 + L%16]
            v2f b;
            b.x = xb[(c * 4 + 2 * hi + 0) * HWN];
            b.y = xb[(c * 4 + 2 * hi + 1) * HWN];
            acc = __builtin_amdgcn_wmma_f32_16x16x4_f32(false, a[c], false, b,
                                                        (short)0, acc, false, false);
        }
#pragma unroll
        for (int r = 0; r < 8; r++)
            og[(size_t)(r + 8 * hi) * HWN + hw0 + col] = acc[r];
    }
}

extern "C" void kernel_launch(void* const* d_in, const int* in_sizes, int n_in,
                              void* d_out, int out_size, void* d_ws, size_t ws_size,
                              hipStream_t stream) {
    (void)in_sizes; (void)n_in; (void)out_size; (void)ws_size;
    const float* x   = (const float*)d_in[0];
    float*       out = (float*)d_out;
    float*       ws  = (float*)d_ws;

    float* gram = ws;           // 4096
    float* sums = ws + 4096;    // 256
    float* W    = ws + 4352;    // 4096
    float* wmu  = ws + 8448;    // 256

    k_zero  <<<17,  256, 0, stream>>>(ws, 4352 + 4096 + 256);
    k_gram  <<<512, 256, 0, stream>>>(x, gram, sums);
    k_chol  <<<1,   16,  0, stream>>>(gram, sums, W, wmu);
    k_whiten<<<512, 256, 0, stream>>>(x, W, wmu, out);
}